// MessageAggregator_46248207843705
// MI455X (gfx1250) — compile-verified
//
#include <hip/hip_runtime.h>
#include <stdint.h>

// Scatter-mean: out[:num_nodes] = arange, out[num_nodes:] = segment_mean(messages, ids)
// Strategy: bin node-ids into 64-node bins (shift, not divide), bucket message
// indices per bin, then one workgroup per bin accumulates rows into LDS with
// conflict-free ds_add_f32 (stride-32 per-lane layout), finally scales by
// 1/max(count,1). Moves the 128M atomic ops from shared L2 to per-WGP LDS.

#define MSG_DIM   128
#define BIN_SHIFT 6
#define BIN_NODES (1 << BIN_SHIFT)   // 64 nodes -> 64*128*4 = 32 KB LDS accumulator
#define BLOCK     256

// ---------------- init: write arange ids, zero bin counters ----------------
__global__ void ma_init_kernel(float* __restrict__ out_ids, int num_nodes,
                               unsigned* __restrict__ bin_counts, int num_bins) {
    int i = blockIdx.x * blockDim.x + threadIdx.x;
    int stride = gridDim.x * blockDim.x;
    for (int k = i; k < num_nodes; k += stride) out_ids[k] = (float)k;
    for (int k = i; k < num_bins;  k += stride) bin_counts[k] = 0u;
}

// ---------------- phase 1: per-bin histogram (LDS-privatized) ----------------
__global__ void ma_bin_count_kernel(const int* __restrict__ ids, int num_msg,
                                    unsigned* __restrict__ bin_counts, int num_bins) {
    __shared__ unsigned h[1024];
    const bool use_lds = (num_bins <= 1024);
    if (use_lds) {
        for (int k = threadIdx.x; k < num_bins; k += blockDim.x) h[k] = 0u;
        __syncthreads();
    }
    int i = blockIdx.x * blockDim.x + threadIdx.x;
    int stride = gridDim.x * blockDim.x;
    for (int m = i; m < num_msg; m += stride) {
        int b = ids[m] >> BIN_SHIFT;
        if (use_lds) atomicAdd(&h[b], 1u);
        else         atomicAdd(&bin_counts[b], 1u);
    }
    if (use_lds) {
        __syncthreads();
        for (int k = threadIdx.x; k < num_bins; k += blockDim.x) {
            unsigned v = h[k];
            if (v) atomicAdd(&bin_counts[k], v);
        }
    }
}

// ---------------- phase 2: exclusive prefix sum over bins (single block) ----
__global__ void ma_scan_kernel(const unsigned* __restrict__ bin_counts,
                               unsigned* __restrict__ offsets,
                               unsigned* __restrict__ cursor, int num_bins) {
    __shared__ unsigned s[1024];
    int tid = threadIdx.x;
    if (num_bins <= 1024) {
        s[tid] = (tid < num_bins) ? bin_counts[tid] : 0u;
        __syncthreads();
        for (int off = 1; off < 1024; off <<= 1) {
            unsigned t = (tid >= off) ? s[tid - off] : 0u;
            __syncthreads();
            s[tid] += t;
            __syncthreads();
        }
        if (tid < num_bins) {
            unsigned excl = (tid == 0) ? 0u : s[tid - 1];
            offsets[tid] = excl;
            cursor[tid]  = excl;
        }
        if (tid == 0) offsets[num_bins] = s[num_bins - 1];
    } else if (tid == 0) {           // generality fallback, not hit for 50K nodes
        unsigned acc = 0;
        for (int k = 0; k < num_bins; ++k) {
            offsets[k] = acc; cursor[k] = acc; acc += bin_counts[k];
        }
        offsets[num_bins] = acc;
    }
}

// ---------------- phase 3: bucket message indices by bin ----------------
__global__ void ma_scatter_idx_kernel(const int* __restrict__ ids, int num_msg,
                                      unsigned* __restrict__ cursor,
                                      unsigned* __restrict__ msg_idx) {
    int i = blockIdx.x * blockDim.x + threadIdx.x;
    int stride = gridDim.x * blockDim.x;
    for (int m = i; m < num_msg; m += stride) {
        int b = ids[m] >> BIN_SHIFT;
        unsigned pos = atomicAdd(&cursor[b], 1u);
        msg_idx[pos] = (unsigned)m;
    }
}

// ---------------- phase 4: per-bin LDS accumulation + mean ----------------
__global__ void __launch_bounds__(BLOCK)
ma_aggregate_kernel(const float* __restrict__ msgs, const int* __restrict__ ids,
                    const unsigned* __restrict__ msg_idx,
                    const unsigned* __restrict__ offsets,
                    float* __restrict__ out_agg, int num_nodes) {
    __shared__ float s_sum[BIN_NODES * MSG_DIM];   // 32768 B
    __shared__ float s_cnt[BIN_NODES];

    const int b = blockIdx.x;
    const int node_base = b << BIN_SHIFT;

    for (int i = threadIdx.x; i < BIN_NODES * MSG_DIM; i += blockDim.x) s_sum[i] = 0.0f;
    for (int i = threadIdx.x; i < BIN_NODES; i += blockDim.x) s_cnt[i] = 0.0f;
    __syncthreads();

    const unsigned start = offsets[b];
    const unsigned end   = offsets[b + 1];
    const int lane   = threadIdx.x & 31;
    // scalarize the wave id so the whole j -> m -> id chain stays in SGPRs
    const int wave   = __builtin_amdgcn_readfirstlane((int)(threadIdx.x >> 5));
    const int nwaves = (int)(blockDim.x >> 5);

    // one wave per message row: 32 lanes x 4 floats at stride 32
    // -> conflict-free ds_add_f32 (lanes hit 32 distinct banks each step)
    for (unsigned j = start + (unsigned)wave; j < end; j += (unsigned)nwaves) {
        const unsigned m = (unsigned)__builtin_amdgcn_readfirstlane(
            (int)__builtin_nontemporal_load(msg_idx + j));
        const int id     = __builtin_amdgcn_readfirstlane(ids[m]);
        const int local  = id - node_base;
        const float* row = msgs + (size_t)m * MSG_DIM;
        float* acc       = s_sum + local * MSG_DIM;
#pragma unroll
        for (int k = 0; k < 4; ++k) {
            const int d = lane + 32 * k;
            float v = __builtin_nontemporal_load(row + d);   // streamed once: TH=NT
            atomicAdd(acc + d, v);                           // ds_add_f32
        }
        if (lane == 0) atomicAdd(&s_cnt[local], 1.0f);
    }
    __syncthreads();

    const int nodes_here = min(BIN_NODES, num_nodes - node_base);
    // turn counts into reciprocals once
    for (int n = threadIdx.x; n < nodes_here; n += blockDim.x)
        s_cnt[n] = 1.0f / fmaxf(s_cnt[n], 1.0f);
    __syncthreads();

    for (int i = threadIdx.x; i < nodes_here * MSG_DIM; i += blockDim.x) {
        const int n = i >> 7;
        out_agg[((size_t)node_base << 7) + i] = s_sum[i] * s_cnt[n];
    }
}

// ---------------- fallback (tiny workspace): direct global atomics ----------
__global__ void ma_fb_init_kernel(float* __restrict__ out, int num_nodes,
                                  float* __restrict__ counts) {
    int i = blockIdx.x * blockDim.x + threadIdx.x;
    int stride = gridDim.x * blockDim.x;
    for (int k = i; k < num_nodes; k += stride) { out[k] = (float)k; counts[k] = 0.0f; }
    float* agg = out + num_nodes;
    size_t total = (size_t)num_nodes * MSG_DIM;
    for (size_t k = i; k < total; k += (size_t)stride) agg[k] = 0.0f;
}

__global__ void ma_fb_scatter_kernel(const float* __restrict__ msgs,
                                     const int* __restrict__ ids, int num_msg,
                                     float* __restrict__ out_agg,
                                     float* __restrict__ counts) {
    const int lane  = threadIdx.x & 31;
    const int gwave = (blockIdx.x * blockDim.x + threadIdx.x) >> 5;
    const int nwave = (gridDim.x * blockDim.x) >> 5;
    for (int m = gwave; m < num_msg; m += nwave) {
        const int id = ids[m];
        const float* row = msgs + (size_t)m * MSG_DIM;
        float* acc = out_agg + (size_t)id * MSG_DIM;
#pragma unroll
        for (int k = 0; k < 4; ++k) {
            const int d = lane + 32 * k;
            atomicAdd(acc + d, __builtin_nontemporal_load(row + d));
        }
        if (lane == 0) atomicAdd(&counts[id], 1.0f);
    }
}

__global__ void ma_fb_final_kernel(float* __restrict__ out_agg,
                                   const float* __restrict__ counts, int num_nodes) {
    size_t total = (size_t)num_nodes * MSG_DIM;
    size_t i = (size_t)blockIdx.x * blockDim.x + threadIdx.x;
    size_t stride = (size_t)gridDim.x * blockDim.x;
    for (size_t k = i; k < total; k += stride)
        out_agg[k] *= 1.0f / fmaxf(counts[k >> 7], 1.0f);
}

// ---------------- host launcher ----------------
extern "C" void kernel_launch(void* const* d_in, const int* in_sizes, int n_in,
                              void* d_out, int out_size, void* d_ws, size_t ws_size,
                              hipStream_t stream) {
    const int*   node_ids = (const int*)d_in[0];
    const float* messages = (const float*)d_in[1];
    const int num_msg   = in_sizes[0];
    const int num_nodes = out_size / (MSG_DIM + 1);   // tuple: ids + agg
    const int num_bins  = (num_nodes + BIN_NODES - 1) >> BIN_SHIFT;

    float* out     = (float*)d_out;
    float* out_agg = out + num_nodes;

    // workspace layout for binned path
    unsigned* bin_counts = (unsigned*)d_ws;
    unsigned* offsets    = bin_counts + num_bins;
    unsigned* cursor     = offsets + num_bins + 1;
    uintptr_t idx_addr   = ((uintptr_t)(cursor + num_bins) + 255u) & ~(uintptr_t)255u;
    unsigned* msg_idx    = (unsigned*)idx_addr;
    size_t binned_bytes  = (idx_addr - (uintptr_t)d_ws) + (size_t)num_msg * sizeof(unsigned);

    const int msg_blocks = min(2048, (num_msg + BLOCK - 1) / BLOCK);

    if (ws_size >= binned_bytes && num_bins >= 1) {
        int init_n = max(num_nodes, num_bins);
        ma_init_kernel<<<(init_n + BLOCK - 1) / BLOCK, BLOCK, 0, stream>>>(
            out, num_nodes, bin_counts, num_bins);
        ma_bin_count_kernel<<<msg_blocks, BLOCK, 0, stream>>>(
            node_ids, num_msg, bin_counts, num_bins);
        ma_scan_kernel<<<1, 1024, 0, stream>>>(bin_counts, offsets, cursor, num_bins);
        ma_scatter_idx_kernel<<<msg_blocks, BLOCK, 0, stream>>>(
            node_ids, num_msg, cursor, msg_idx);
        ma_aggregate_kernel<<<num_bins, BLOCK, 0, stream>>>(
            messages, node_ids, msg_idx, offsets, out_agg, num_nodes);
    } else {
        // small-workspace fallback: direct global atomics (counts only in ws)
        float* counts = (float*)d_ws;
        size_t total = (size_t)num_nodes * MSG_DIM;
        int zb = (int)min((size_t)4096, (total + BLOCK - 1) / BLOCK);
        ma_fb_init_kernel<<<zb, BLOCK, 0, stream>>>(out, num_nodes, counts);
        ma_fb_scatter_kernel<<<msg_blocks, BLOCK, 0, stream>>>(
            messages, node_ids, num_msg, out_agg, counts);
        ma_fb_final_kernel<<<zb, BLOCK, 0, stream>>>(out_agg, counts, num_nodes);
    }
}